// WeightedMSELoss_6116033429792
// MI455X (gfx1250) — compile-verified
//
#include <hip/hip_runtime.h>
#include <hip/hip_bf16.h>

typedef __attribute__((ext_vector_type(2))) float v2f;
typedef __attribute__((ext_vector_type(4))) float f32x4;
typedef __attribute__((ext_vector_type(8))) float v8f;

// Reproduces searchsorted(EDGES, t, 'left')-1 binning exactly:
// bin i <=> EDGES[i] < t <= EDGES[i+1]; weight = 2^bin; out-of-range -> 0.
__device__ __forceinline__ float bin_weight(float t) {
    int b = (int)(t > 0.2f) + (int)(t > 0.4f) + (int)(t > 0.6f) + (int)(t > 0.8f);
    float w = (float)(1 << b);
    return ((t > 0.0f) && (t <= 1.0f)) ? w : 0.0f;
}

// Cross-lane (wave32) sum using the matrix engine.
// A is 16x4 f32 striped as: lane L, VGPR0 -> A[L%16][L<16?0:2]; VGPR1 -> K=1/3 (set 0).
// With B = ones(4x16): D[m][n] = v_m + v_{m+16}. Lanes 0-15 hold D rows 0..7 in
// c[0..7], lanes 16-31 hold rows 8..15. Per-lane sum of c[] then one xor-16
// shuffle yields the full 32-lane sum in every lane.
__device__ __forceinline__ float wave_reduce_wmma(float v) {
    v2f a; a[0] = v;    a[1] = 0.0f;
    v2f b; b[0] = 1.0f; b[1] = 1.0f;
    v8f c = {};
    c = __builtin_amdgcn_wmma_f32_16x16x4_f32(false, a, false, b,
                                              (short)0, c, false, false);
    float s = ((c[0] + c[1]) + (c[2] + c[3])) + ((c[4] + c[5]) + (c[6] + c[7]));
    s += __shfl_xor(s, 16, 32);
    return s;
}

// Full-block reduction; result valid in thread 0. Deterministic (fixed order).
__device__ __forceinline__ float block_reduce(float v, float* lds) {
    float s = wave_reduce_wmma(v);
    const int wid  = threadIdx.x >> 5;
    const int lane = threadIdx.x & 31;
    if (lane == 0) lds[wid] = s;
    __syncthreads();
    float r = 0.0f;
    if (threadIdx.x == 0) {
        const int nw = blockDim.x >> 5;
        for (int i = 0; i < nw; ++i) r += lds[i];
    }
    return r;
}

__global__ void wmse_partial_kernel(const float* __restrict__ pred,
                                    const float* __restrict__ targ,
                                    float* __restrict__ partial,
                                    int nvec4, int n) {
    __shared__ float lds[8];
    const f32x4* __restrict__ p4 = (const f32x4*)pred;
    const f32x4* __restrict__ t4 = (const f32x4*)targ;

    float acc = 0.0f;
    const int stride = gridDim.x * blockDim.x;
    int i = blockIdx.x * blockDim.x + threadIdx.x;

    // Streaming main loop: 16B non-temporal loads (read-once data, bypass cache
    // residency) -> global_load_b128 th:NT. 512B/wave/load, fully coalesced.
    for (; i < nvec4; i += stride) {
        f32x4 p = __builtin_nontemporal_load(&p4[i]);
        f32x4 t = __builtin_nontemporal_load(&t4[i]);
#pragma unroll
        for (int k = 0; k < 4; ++k) {
            float d = p[k] - t[k];
            acc = fmaf(bin_weight(t[k]) * d, d, acc);
        }
    }
    // Tail (N % 4) — N = 2^25 so normally empty, kept for generality.
    for (int j = nvec4 * 4 + blockIdx.x * blockDim.x + threadIdx.x; j < n; j += stride) {
        float d = pred[j] - targ[j];
        acc = fmaf(bin_weight(targ[j]) * d, d, acc);
    }

    float bs = block_reduce(acc, lds);
    if (threadIdx.x == 0) partial[blockIdx.x] = bs;
}

__global__ void wmse_final_kernel(const float* __restrict__ partial, int n,
                                  float* __restrict__ out) {
    __shared__ float lds[8];
    float acc = 0.0f;
    for (int i = threadIdx.x; i < n; i += blockDim.x) acc += partial[i];
    float s = block_reduce(acc, lds);
    if (threadIdx.x == 0) out[0] = s / 31.0f;  // sum(WEIGHTS) = 31
}

extern "C" void kernel_launch(void* const* d_in, const int* in_sizes, int n_in,
                              void* d_out, int out_size, void* d_ws, size_t ws_size,
                              hipStream_t stream) {
    const float* pred = (const float*)d_in[0];
    const float* targ = (const float*)d_in[1];
    float* out = (float*)d_out;
    float* partial = (float*)d_ws;

    const int n = in_sizes[0];
    const int nvec4 = n / 4;

    // 2048 blocks x 256 threads = 512K threads = 16K waves: enough MLP to pin
    // HBM at ~23 TB/s; each thread does 16 x (2 x b128 NT loads). Capped by
    // available scratch (one f32 partial per block); ws_size is fixed per run
    // so the launch shape is deterministic.
    int blocks = 2048;
    const size_t max_partials = ws_size / sizeof(float);
    if ((size_t)blocks > max_partials) blocks = (int)max_partials;
    if (blocks < 1) blocks = 1;

    wmse_partial_kernel<<<blocks, 256, 0, stream>>>(pred, targ, partial, nvec4, n);
    wmse_final_kernel<<<1, 256, 0, stream>>>(partial, blocks, out);
}